// SymPlaneLoss_18760417149453
// MI455X (gfx1250) — compile-verified
//
#include <hip/hip_runtime.h>
#include <math.h>

typedef float v2f __attribute__((ext_vector_type(2)));
typedef float v8f __attribute__((ext_vector_type(8)));

#define GDIM 64
#define GG3  (GDIM*GDIM*GDIM)   // 262144
#define NPTS 65536
#define NPL  12
#define NB   8

__global__ void sym_init_out(float* out) {
    if (threadIdx.x == 0 && blockIdx.x == 0) out[0] = 0.0f;
}

__global__ __launch_bounds__(256) void sym_plane_loss_kernel(
    const float* __restrict__ voxel,        // (B, G^3) as (B,G,G,G)
    const float* __restrict__ points,       // (B, N, 3)
    const float* __restrict__ cpnts,        // (B, G^3, 3)
    const float* __restrict__ planes,       // (B, P, 4)
    float* __restrict__ out)
{
    __shared__ float sp[NB * NPL * 4];      // normalized planes: nx,ny,nz,d
    const int tid = threadIdx.x;
    if (tid < NB * NPL) {
        float nx = planes[tid*4+0], ny = planes[tid*4+1];
        float nz = planes[tid*4+2], dd = planes[tid*4+3];
        float inv = 1.0f / sqrtf(nx*nx + ny*ny + nz*nz);
        sp[tid*4+0] = nx*inv; sp[tid*4+1] = ny*inv;
        sp[tid*4+2] = nz*inv; sp[tid*4+3] = dd*inv;
    }
    __syncthreads();

    const int total  = NB * NPTS;           // 524288 points
    const int stride = gridDim.x * blockDim.x;
    float acc = 0.0f;

    for (int i = blockIdx.x * blockDim.x + tid; i < total; i += stride) {
        const int b = i >> 16;              // NPTS == 65536
        const int n = i & (NPTS - 1);

        const float* pp = points + (size_t)i * 3;
        const float px = pp[0], py = pp[1], pz = pp[2];

        // flat j range for this point: [12n, 12n+12); plane = j >> 16
        const int j0 = n * NPL;
        const int p0 = j0 >> 16;
        const int p1 = (j0 + NPL - 1) >> 16;
        const int c0 = (p0 == p1) ? NPL : ((p1 << 16) - j0); // reps on plane p0

        for (int t = 0; t < 2; ++t) {
            if (t && (p0 == p1)) break;
            const int   p = t ? p1 : p0;
            const float w = (float)(t ? (NPL - c0) : c0);

            const float* pl = &sp[(b * NPL + p) * 4];
            const float nx = pl[0], ny = pl[1], nz = pl[2], dd = pl[3];

            const float q  = 2.0f * (px*nx + py*ny + pz*nz + dd);
            const float rx = px - q*nx, ry = py - q*ny, rz = pz - q*nz;

            int ix = (int)ceilf((rx + 0.5f) * (float)GDIM - 0.5f);
            int iy = (int)ceilf((ry + 0.5f) * (float)GDIM - 0.5f);
            int iz = (int)ceilf((rz + 0.5f) * (float)GDIM - 0.5f);
            int idx = ix * (GDIM*GDIM) + iy * GDIM + iz;     // clip combined idx (matches ref)
            idx = idx < 0 ? 0 : (idx > (GG3 - 1) ? (GG3 - 1) : idx);

            const size_t base = (size_t)b * GG3 + (size_t)idx;
            const float* cp = cpnts + base * 3;
            const float  m  = 1.0f - voxel[base];
            const float dx = rx - cp[0], dy = ry - cp[1], dz = rz - cp[2];
            acc += w * (m * m) * (dx*dx + dy*dy + dz*dz);
        }
    }

    // ---- wave32 reduction via V_WMMA_F32_16X16X4_F32 ----
    // A: VGPR0 = per-lane partial -> A[m][0]=v[m] (lanes 0-15), A[m][2]=v[m+16] (lanes 16-31)
    //    VGPR1 = 0 -> A[m][1]=A[m][3]=0
    // B: all ones (layout-independent).  D[m][n] = v[m] + v[m+16] for every n.
    // Sum of the 8 D regs: lanes 0-15 hold sum(v[0..7])+sum(v[16..23]),
    // lanes 16-31 hold sum(v[8..15])+sum(v[24..31]); xor-16 add gives the full sum.
    v2f a;  a.x = acc;  a.y = 0.0f;
    v2f bo; bo.x = 1.0f; bo.y = 1.0f;
    v8f c = {};
    v8f d = __builtin_amdgcn_wmma_f32_16x16x4_f32(
        /*neg_a=*/false, a, /*neg_b=*/false, bo,
        /*c_mod=*/(short)0, c, /*reuse_a=*/false, /*reuse_b=*/false);
    float s = d[0] + d[1] + d[2] + d[3] + d[4] + d[5] + d[6] + d[7];
    s += __shfl_xor(s, 16, 32);

    if ((tid & 31) == 0) {
        atomicAdd(out, s * (1.0f / 96.0f));   // mean over (B,P) = /96
    }
}

extern "C" void kernel_launch(void* const* d_in, const int* in_sizes, int n_in,
                              void* d_out, int out_size, void* d_ws, size_t ws_size,
                              hipStream_t stream) {
    (void)in_sizes; (void)n_in; (void)out_size; (void)d_ws; (void)ws_size;
    const float* voxel  = (const float*)d_in[0];
    const float* points = (const float*)d_in[1];
    const float* cpnts  = (const float*)d_in[2];
    const float* planes = (const float*)d_in[3];
    float* out = (float*)d_out;

    sym_init_out<<<1, 32, 0, stream>>>(out);
    // 512 blocks x 256 threads = 131072 threads; 4 points each (524288 total), full waves
    sym_plane_loss_kernel<<<512, 256, 0, stream>>>(voxel, points, cpnts, planes, out);
}